// EntityPairRepresentationMultiInstance_55198919688619
// MI455X (gfx1250) — compile-verified
//
#include <hip/hip_runtime.h>
#include <hip/hip_bf16.h>

// ---- problem constants (match reference) ----
#define B_     2
#define E_     300
#define M_     100
#define T_     256
#define P_     200
#define MP_    1024
#define K_     64
#define H_     256
#define META_  32
#define MEM_   64
#define F1_    1344   // 5*H + 2*META
#define F2_    384    // H + 2*MEM
#define NEGV  (-1e30f)

typedef __attribute__((ext_vector_type(2))) float v2f;
typedef __attribute__((ext_vector_type(8))) float v8f;

// ---------------------------------------------------------------------------
// K1: build X1[b*MP+mp][1344] = [ctx(256) | mpr(512) | epg(512) | tok(32) | sent(32)]
// one block per (b,mp); 256 threads, one per H dim.
// ---------------------------------------------------------------------------
__global__ void build_features_kernel(
    const float* __restrict__ h,              // [B,T,H]
    const int*   __restrict__ ctx_masks,      // [B,MP,T]
    const float* __restrict__ mention_reprs,  // [B,M,H]
    const int*   __restrict__ rel_mention_pairs, // [B,MP,2]
    const int*   __restrict__ rel_mention_pair_ep, // [B,MP]
    const int*   __restrict__ pairs,          // [B,P,2]
    const float* __restrict__ entity_reprs,   // [B,E,H]
    const int*   __restrict__ tok_dist,       // [B,MP]
    const int*   __restrict__ sent_dist,      // [B,MP]
    const float* __restrict__ tok_emb,        // [TOKD,META]
    const float* __restrict__ sent_emb,       // [SENTD,META]
    float*       __restrict__ X)              // [B*MP, F1]
{
    const int bmp = blockIdx.x;           // 0 .. B*MP-1
    const int b   = bmp / MP_;
    const int mp  = bmp % MP_;
    const int tid = threadIdx.x;          // 0..255

    __shared__ int smask[T_];
    smask[tid] = ctx_masks[(size_t)bmp * T_ + tid];
    __syncthreads();

    // masked max-pool over tokens; mask value is uniform across the block per t
    const float* hb = h + (size_t)b * T_ * H_;
    float mx = NEGV;
    int any = 0;
    #pragma unroll 4
    for (int t = 0; t < T_; ++t) {
        if (smask[t]) {
            any = 1;
            mx = fmaxf(mx, hb[(size_t)t * H_ + tid]);
        }
    }

    float* xr = X + (size_t)bmp * F1_;
    xr[tid] = any ? mx : 0.0f;

    // mention pair gather
    const int m0 = rel_mention_pairs[(size_t)bmp * 2 + 0];
    const int m1 = rel_mention_pairs[(size_t)bmp * 2 + 1];
    xr[H_     + tid] = mention_reprs[((size_t)b * M_ + m0) * H_ + tid];
    xr[2*H_   + tid] = mention_reprs[((size_t)b * M_ + m1) * H_ + tid];

    // entity pair gather (through rel_mention_pair_ep -> pairs)
    const int p  = rel_mention_pair_ep[bmp];
    const int e0 = pairs[((size_t)b * P_ + p) * 2 + 0];
    const int e1 = pairs[((size_t)b * P_ + p) * 2 + 1];
    xr[3*H_   + tid] = entity_reprs[((size_t)b * E_ + e0) * H_ + tid];
    xr[4*H_   + tid] = entity_reprs[((size_t)b * E_ + e1) * H_ + tid];

    // distance embeddings (32 dims each)
    if (tid < META_) {
        const int td = tok_dist[bmp];
        const int sd = sent_dist[bmp];
        xr[5*H_          + tid] = tok_emb [(size_t)td * META_ + tid];
        xr[5*H_ + META_  + tid] = sent_emb[(size_t)sd * META_ + tid];
    }
}

// ---------------------------------------------------------------------------
// WMMA f32 GEMM: C[Mrows x Ncols] = A[Mrows x Kd] * W^T + bias, optional ReLU
// W is row-major [Ncols x Kd] (i.e. B[k][n] = W[n*Kd + k]).
// One wave per 16x16 C tile. Mrows, Ncols multiples of 16; Kd multiple of 4.
// Uses V_WMMA_F32_16X16X4_F32.
// ---------------------------------------------------------------------------
__global__ void wmma_gemm_bias_kernel(
    const float* __restrict__ A,
    const float* __restrict__ W,
    const float* __restrict__ bias,
    float*       __restrict__ C,
    int Mrows, int Ncols, int Kd, int do_relu)
{
    const int wavesPerBlock = blockDim.x >> 5;
    const int wave = (int)(threadIdx.x >> 5);
    const int lane = (int)(threadIdx.x & 31);
    const int tileId = blockIdx.x * wavesPerBlock + wave;
    const int tilesN = Ncols >> 4;
    const int totalTiles = (Mrows >> 4) * tilesN;
    if (tileId >= totalTiles) return;     // whole-wave uniform exit

    const int tm = tileId / tilesN;
    const int tn = tileId % tilesN;
    const int m0 = tm << 4;
    const int n0 = tn << 4;

    const int half = lane >> 4;           // 0: K={0,1}; 1: K={2,3}
    const int l    = lane & 15;

    // A frag: lane row = m0+l (both halves), K offset = 2*half
    const float* arow = A + (size_t)(m0 + l) * Kd + 2 * half;
    // B frag: lane col = n0+l, same K offset pattern
    const float* wrow = W + (size_t)(n0 + l) * Kd + 2 * half;

    v8f c = {};
    for (int k = 0; k < Kd; k += 4) {
        v2f a = *(const v2f*)(arow + k);
        v2f bfrag = *(const v2f*)(wrow + k);
        c = __builtin_amdgcn_wmma_f32_16x16x4_f32(
                /*neg_a=*/false, a, /*neg_b=*/false, bfrag,
                /*c_mod=*/(short)0, c, /*reuse_a=*/false, /*reuse_b=*/false);
    }

    // C/D layout: element r -> row m0 + r + 8*half, col n0 + l
    const float bv = bias[n0 + l];
    #pragma unroll
    for (int r = 0; r < 8; ++r) {
        float v = c[r] + bv;
        if (do_relu) v = fmaxf(v, 0.0f);
        C[(size_t)(m0 + r + 8 * half) * Ncols + (n0 + l)] = v;
    }
}

// ---------------------------------------------------------------------------
// K3: build X2[b*P+p][384] = [ g(256) | et0(64) | et1(64) ]
// g[d] = max_k ( local[b, idx_k, d] + (mask_k ? 0 : NEG) )
// one block per (b,p); 256 threads.
// ---------------------------------------------------------------------------
__global__ void build_g_kernel(
    const float* __restrict__ local,            // [B*MP, H]
    const int*   __restrict__ rel_entity_pair_mp, // [B,P,K]
    const int*   __restrict__ rel_pair_masks,   // [B,P,K]
    const int*   __restrict__ rel_entity_types, // [B,P,2]
    const float* __restrict__ entity_memory,    // [NTYPES, MEM]
    float*       __restrict__ X2)               // [B*P, F2]
{
    const int bp  = blockIdx.x;       // 0 .. B*P-1
    const int b   = bp / P_;
    const int tid = threadIdx.x;      // 0..255

    __shared__ int sidx[K_];
    __shared__ int smask[K_];
    if (tid < K_) {
        sidx [tid] = rel_entity_pair_mp[(size_t)bp * K_ + tid];
        smask[tid] = rel_pair_masks   [(size_t)bp * K_ + tid];
    }
    __syncthreads();

    const float* lb = local + (size_t)b * MP_ * H_;
    float mx = -__builtin_inff();
    #pragma unroll 4
    for (int k = 0; k < K_; ++k) {
        float v = lb[(size_t)sidx[k] * H_ + tid] + (smask[k] ? 0.0f : NEGV);
        mx = fmaxf(mx, v);
    }

    float* xr = X2 + (size_t)bp * F2_;
    xr[tid] = mx;

    if (tid < 2 * MEM_) {
        const int which = tid / MEM_;
        const int t = rel_entity_types[(size_t)bp * 2 + which];
        xr[H_ + tid] = entity_memory[(size_t)t * MEM_ + (tid % MEM_)];
    }
}

// ---------------------------------------------------------------------------
extern "C" void kernel_launch(void* const* d_in, const int* in_sizes, int n_in,
                              void* d_out, int out_size, void* d_ws, size_t ws_size,
                              hipStream_t stream) {
    const int*   pairs               = (const int*)  d_in[0];
    const float* entity_reprs        = (const float*)d_in[1];
    const float* h                   = (const float*)d_in[2];
    const float* mention_reprs       = (const float*)d_in[3];
    const float* entity_memory       = (const float*)d_in[4];
    const int*   rel_entity_pair_mp  = (const int*)  d_in[5];
    const int*   rel_mention_pair_ep = (const int*)  d_in[6];
    const int*   rel_mention_pairs   = (const int*)  d_in[7];
    const int*   rel_ctx_masks       = (const int*)  d_in[8];
    const int*   rel_pair_masks      = (const int*)  d_in[9];
    const int*   rel_token_distances = (const int*)  d_in[10];
    const int*   rel_sentence_dist   = (const int*)  d_in[11];
    const int*   rel_entity_types    = (const int*)  d_in[12];
    const float* pair_w              = (const float*)d_in[13]; // [H, F1]
    const float* pair_b              = (const float*)d_in[14]; // [H]
    const float* rel_w               = (const float*)d_in[15]; // [H, F2]
    const float* rel_b               = (const float*)d_in[16]; // [H]
    const float* tok_emb             = (const float*)d_in[17];
    const float* sent_emb            = (const float*)d_in[18];
    float* out = (float*)d_out;                                 // [B*P, H]

    // workspace layout (floats)
    float* X1    = (float*)d_ws;                         // [B*MP, F1]
    float* local = X1    + (size_t)B_ * MP_ * F1_;       // [B*MP, H]
    float* X2    = local + (size_t)B_ * MP_ * H_;        // [B*P, F2]

    // K1: feature build — one block per mention pair
    build_features_kernel<<<B_ * MP_, 256, 0, stream>>>(
        h, rel_ctx_masks, mention_reprs, rel_mention_pairs, rel_mention_pair_ep,
        pairs, entity_reprs, rel_token_distances, rel_sentence_dist,
        tok_emb, sent_emb, X1);

    // K2: local = X1 @ pair_w^T + pair_b   (2048x256, Kd=1344)
    {
        const int tiles = (B_ * MP_ / 16) * (H_ / 16);   // 2048 tiles
        const int wavesPerBlock = 8;                     // 256 threads
        const int blocks = (tiles + wavesPerBlock - 1) / wavesPerBlock;
        wmma_gemm_bias_kernel<<<blocks, 32 * wavesPerBlock, 0, stream>>>(
            X1, pair_w, pair_b, local, B_ * MP_, H_, F1_, /*relu=*/0);
    }

    // K3: masked K-max + entity-memory concat — one block per entity pair
    build_g_kernel<<<B_ * P_, 256, 0, stream>>>(
        local, rel_entity_pair_mp, rel_pair_masks, rel_entity_types,
        entity_memory, X2);

    // K4: out = relu(X2 @ rel_w^T + rel_b)   (400x256, Kd=384)
    {
        const int tiles = (B_ * P_ / 16) * (H_ / 16);    // 400 tiles
        const int wavesPerBlock = 8;
        const int blocks = (tiles + wavesPerBlock - 1) / wavesPerBlock;
        wmma_gemm_bias_kernel<<<blocks, 32 * wavesPerBlock, 0, stream>>>(
            X2, rel_w, rel_b, out, B_ * P_, H_, F2_, /*relu=*/1);
    }
}